// NegSTOILoss_29343216566897
// MI455X (gfx1250) — compile-verified
//
#include <hip/hip_runtime.h>
#include <math.h>

// ---------------- problem constants (match reference) ----------------
static constexpr int   BATCH  = 32;
static constexpr int   T      = 320000;
static constexpr int   WINL   = 256;
static constexpr int   HOP    = 128;
static constexpr int   NF     = 2499;           // frames of the raw signal
static constexpr int   NFR    = 2498;           // STFT frames (nf-1)
static constexpr int   NBINS  = 257;            // rfft bins of 512-pt FFT
static constexpr int   NPAD   = 272;            // bins padded to 17*16
static constexpr int   WCOLS  = 544;            // [Re 272 | Im 272]
static constexpr int   NBAND  = 15;
static constexpr int   NSEG   = 30;             // N_INTEL
static constexpr int   M      = NFR - NSEG + 1; // 2469 segments
static constexpr int   FT     = (NFR + 15) / 16;// 157 frame tiles
static constexpr float EPSF   = 1e-8f;
static constexpr float CLIP1  = 1.0f + 5.623413251903491f; // 1 + 10^(0.75)

typedef float v2f __attribute__((ext_vector_type(2)));
typedef float v8f __attribute__((ext_vector_type(8)));

// ---------------- 1) per-frame windowed norms ----------------
__global__ __launch_bounds__(256) void k_energy(const float* __restrict__ tgt,
                                                const float* __restrict__ win,
                                                float* __restrict__ norms) {
  int frame = blockIdx.x * 8 + (threadIdx.x >> 5);
  int lane  = threadIdx.x & 31;
  if (frame >= BATCH * NF) return;
  int b = frame / NF, n = frame % NF;
  const float* p = tgt + (size_t)b * T + (size_t)n * HOP;
  float s = 0.f;
  for (int i = lane; i < WINL; i += 32) { float v = p[i] * win[i]; s += v * v; }
  for (int o = 16; o; o >>= 1) s += __shfl_xor(s, o, 32);
  if (lane == 0) norms[frame] = sqrtf(s);
}

// ---------------- 2) per-batch max + stable compaction ----------------
__global__ __launch_bounds__(1024) void k_scan(const float* __restrict__ norms,
                                               int* __restrict__ order,
                                               int* __restrict__ counts) {
  int b = blockIdx.x;
  const float* nr = norms + (size_t)b * NF;
  __shared__ float smax[1024];
  __shared__ int   scnt[1024];
  int t = threadIdx.x;
  float mx = -1e30f;
  for (int i = t; i < NF; i += 1024) mx = fmaxf(mx, nr[i]);
  smax[t] = mx; __syncthreads();
  for (int s = 512; s; s >>= 1) { if (t < s) smax[t] = fmaxf(smax[t], smax[t + s]); __syncthreads(); }
  // keep frame iff 20log10(norm+eps) > 20log10(max+eps) - 40  <=>  norm+eps > (max+eps)/100
  float thr = (smax[0] + EPSF) * 0.01f;
  __syncthreads();
  const int CH = 3; // ceil(2499/1024)
  int base = t * CH, local = 0;
  for (int i = 0; i < CH; i++) { int idx = base + i; if (idx < NF && nr[idx] + EPSF > thr) local++; }
  scnt[t] = local; __syncthreads();
  for (int off = 1; off < 1024; off <<= 1) {
    int v = scnt[t]; int add = (t >= off) ? scnt[t - off] : 0;
    __syncthreads(); scnt[t] = v + add; __syncthreads();
  }
  int w = scnt[t] - local;
  for (int i = 0; i < CH; i++) {
    int idx = base + i;
    if (idx < NF && nr[idx] + EPSF > thr) order[(size_t)b * NF + (w++)] = idx;
  }
  if (t == 1023) counts[b] = scnt[1023];
}

// ---------------- 3) DFT matrix W[k][WCOLS] + bin->band map ----------------
__global__ __launch_bounds__(256) void k_prep(const float* __restrict__ obm,
                                              float* __restrict__ W,
                                              int* __restrict__ bandmap) {
  int i = blockIdx.x * 256 + threadIdx.x;
  if (i < WCOLS * WINL) {
    int k = i / WCOLS, n = i % WCOLS;
    float v = 0.f;
    if (n < NBINS) {                       // Re: cos(2*pi*f*k/512)
      int p = (n * k) & 511;
      v = cosf(6.2831853071795864f * (float)p / 512.f);
    } else if (n >= NPAD && n < NPAD + NBINS) { // Im: -sin(2*pi*f*k/512)
      int f = n - NPAD;
      int p = (f * k) & 511;
      v = -sinf(6.2831853071795864f * (float)p / 512.f);
    }
    W[i] = v;
  } else if (i < WCOLS * WINL + NPAD) {
    int f = i - WCOLS * WINL;
    int band = -1;
    if (f < NBINS) for (int k = 0; k < NBAND; k++) if (obm[k * NBINS + f] > 0.5f) band = k;
    bandmap[f] = band;
  }
}

// ---------------- 4) OLA + STFT (WMMA f32 GEMM) + band energies ----------------
// D = A(16 frames x 256) * W(256 x 544).  One wave owns a (Re,Im) 16x16 tile pair.
__global__ __launch_bounds__(256) void k_dft(const float* __restrict__ est,
                                             const float* __restrict__ tgt,
                                             const float* __restrict__ win,
                                             const float* __restrict__ W,
                                             const int* __restrict__ bandmap,
                                             const int* __restrict__ order,
                                             const int* __restrict__ counts,
                                             float* __restrict__ tob) {
  int tile = blockIdx.x;            // frame tile 0..FT-1
  int b    = blockIdx.y;            // batch
  int sig  = blockIdx.z;            // 0 = target(x), 1 = estimate(y)
  const float* src = (sig ? est : tgt) + (size_t)b * T;
  int cnt = counts[b];
  const int* ord = order + (size_t)b * NF;

  __shared__ float A[16][260];          // padded rows: 260 % 64 = 4 -> conflict-free column reads
  __shared__ float Esh[17][16][17];     // per-tile |X|^2, padded
  __shared__ int   sBand[NPAD];

  // stage band map
  for (int i = threadIdx.x; i < NPAD; i += 256) sBand[i] = bandmap[i];

  // stage A: on-the-fly overlap-add of compacted windowed frames, re-windowed for STFT
  for (int e = threadIdx.x; e < 16 * WINL; e += 256) {
    int m = e >> 8, k = e & 255;
    int n = tile * 16 + m;
    float v = 0.f;
    if (n < NFR) {
      int t0 = n * HOP + k;
      int j = t0 >> 7, r = t0 & 127;
      float s = 0.f;
      if (j < cnt)               s += src[(size_t)ord[j] * HOP + r] * win[r];
      if (j >= 1 && j - 1 < cnt) s += src[(size_t)ord[j - 1] * HOP + HOP + r] * win[HOP + r];
      v = s * win[k];
    }
    A[m][k] = v;
  }
  __syncthreads();

  int wave = threadIdx.x >> 5, lane = threadIdx.x & 31;
  int mrow  = lane & 15;
  int khalf = (lane >> 4) * 2;      // lanes 0-15: K+{0,1}; lanes 16-31: K+{2,3}

  for (int p = wave; p < 17; p += 8) {      // 17 (Re,Im) column-tile pairs
    v8f accR = {};
    v8f accI = {};
    int nR = p * 16 + (lane & 15);
    int nI = nR + NPAD;
    for (int k0 = 0; k0 < WINL; k0 += 4) {
      v2f a;  a.x = A[mrow][k0 + khalf];  a.y = A[mrow][k0 + khalf + 1];
      const float* wr = W + (size_t)(k0 + khalf) * WCOLS;
      v2f bR; bR.x = wr[nR]; bR.y = wr[WCOLS + nR];
      v2f bI; bI.x = wr[nI]; bI.y = wr[WCOLS + nI];
      accR = __builtin_amdgcn_wmma_f32_16x16x4_f32(false, a, false, bR, (short)0, accR, false, false);
      accI = __builtin_amdgcn_wmma_f32_16x16x4_f32(false, a, false, bI, (short)0, accI, false, false);
    }
    int fcol  = lane & 15;
    int mbase = (lane < 16) ? 0 : 8;   // C/D layout: vgpr r -> M = r (+8 for hi lanes)
#pragma unroll
    for (int r = 0; r < 8; r++) {
      float e = accR[r] * accR[r] + accI[r] * accI[r];
      Esh[p][mbase + r][fcol] = e;
    }
  }
  __syncthreads();

  // deterministic band reduction: thread = (frame m, band)
  if (threadIdx.x < 16 * NBAND) {
    int m = threadIdx.x / NBAND, band = threadIdx.x % NBAND;
    float s = 0.f;
    for (int f = 0; f < NBINS; f++)
      if (sBand[f] == band) s += Esh[f >> 4][m][f & 15];
    int n = tile * 16 + m;
    if (n < NFR)
      tob[(((size_t)sig * BATCH + b) * NBAND + band) * NFR + n] = sqrtf(s + EPSF);
  }
}

// ---------------- 5) segment correlations (one wave per (b,mpos)) ----------------
__global__ __launch_bounds__(256) void k_corr(const float* __restrict__ tob,
                                              const int* __restrict__ counts,
                                              float* __restrict__ corrbuf) {
  int gw   = (int)((blockIdx.x * 256 + threadIdx.x) >> 5);
  int lane = threadIdx.x & 31;
  if (gw >= BATCH * M) return;
  int b = gw / M, mpos = gw % M;
  int cnt = counts[b];
  float val = 0.f;
  if (mpos < cnt - NSEG && lane < NBAND) {
    const float* xr = tob + (((size_t)0 * BATCH + b) * NBAND + lane) * NFR + mpos;
    const float* yr = tob + (((size_t)1 * BATCH + b) * NBAND + lane) * NFR + mpos;
    float x[NSEG], y[NSEG];
    float sx2 = 0.f, sy2 = 0.f;
#pragma unroll
    for (int j = 0; j < NSEG; j++) { x[j] = xr[j]; y[j] = yr[j]; sx2 += x[j] * x[j]; sy2 += y[j] * y[j]; }
    float nc = sqrtf(sx2) / (sqrtf(sy2) + EPSF);
    float sy = 0.f, sx = 0.f;
#pragma unroll
    for (int j = 0; j < NSEG; j++) { y[j] = fminf(y[j] * nc, x[j] * CLIP1); sy += y[j]; sx += x[j]; }
    float my = sy * (1.f / NSEG), mx = sx * (1.f / NSEG);
    float ny2 = 0.f, nx2 = 0.f, dot = 0.f;
#pragma unroll
    for (int j = 0; j < NSEG; j++) {
      float yc = y[j] - my, xc = x[j] - mx;
      ny2 += yc * yc; nx2 += xc * xc; dot += yc * xc;
    }
    val = dot / ((sqrtf(ny2) + EPSF) * (sqrtf(nx2) + EPSF));
  }
  for (int o = 16; o; o >>= 1) val += __shfl_xor(val, o, 32);
  if (lane == 0) corrbuf[gw] = val * (1.f / NBAND);
}

// ---------------- 6) per-batch final reduction ----------------
__global__ __launch_bounds__(256) void k_final(const float* __restrict__ corrbuf,
                                               const int* __restrict__ counts,
                                               float* __restrict__ out) {
  int b = blockIdx.x;
  __shared__ float s[256];
  float acc = 0.f;
  for (int i = threadIdx.x; i < M; i += 256) acc += corrbuf[(size_t)b * M + i];
  s[threadIdx.x] = acc; __syncthreads();
  for (int o = 128; o; o >>= 1) { if (threadIdx.x < o) s[threadIdx.x] += s[threadIdx.x + o]; __syncthreads(); }
  if (threadIdx.x == 0) {
    int V = counts[b] - NSEG;
    if (V < 0) V = 0; if (V > M) V = M;
    out[b] = -s[0] / ((float)V + EPSF);
  }
}

// ---------------- launcher ----------------
extern "C" void kernel_launch(void* const* d_in, const int* in_sizes, int n_in,
                              void* d_out, int out_size, void* d_ws, size_t ws_size,
                              hipStream_t stream) {
  (void)in_sizes; (void)n_in; (void)out_size; (void)ws_size;
  const float* est = (const float*)d_in[0];
  const float* tgt = (const float*)d_in[1];
  const float* win = (const float*)d_in[2];
  const float* obm = (const float*)d_in[3];
  float* out = (float*)d_out;

  char* base = (char*)d_ws;
  size_t off = 0;
  auto alloc = [&](size_t bytes) -> char* {
    size_t o = (off + 255) & ~(size_t)255;
    off = o + bytes;
    return base + o;
  };
  float* norms   = (float*)alloc((size_t)BATCH * NF * 4);
  int*   order   = (int*)  alloc((size_t)BATCH * NF * 4);
  int*   counts  = (int*)  alloc((size_t)BATCH * 4);
  float* W       = (float*)alloc((size_t)WCOLS * WINL * 4);
  int*   bandmap = (int*)  alloc((size_t)NPAD * 4);
  float* tob     = (float*)alloc((size_t)2 * BATCH * NBAND * NFR * 4);
  float* corrbuf = (float*)alloc((size_t)BATCH * M * 4);

  k_energy<<<(BATCH * NF + 7) / 8, 256, 0, stream>>>(tgt, win, norms);
  k_scan<<<BATCH, 1024, 0, stream>>>(norms, order, counts);
  k_prep<<<(WCOLS * WINL + NPAD + 255) / 256, 256, 0, stream>>>(obm, W, bandmap);
  k_dft<<<dim3(FT, BATCH, 2), 256, 0, stream>>>(est, tgt, win, W, bandmap, order, counts, tob);
  k_corr<<<((BATCH * M) * 32 + 255) / 256, 256, 0, stream>>>(tob, counts, corrbuf);
  k_final<<<BATCH, 256, 0, stream>>>(corrbuf, counts, out);
}